// GraphBackboneGAT_40956808135087
// MI455X (gfx1250) — compile-verified
//
#include <hip/hip_runtime.h>

typedef __attribute__((ext_vector_type(2))) float v2f;
typedef __attribute__((ext_vector_type(4))) float v4f;
typedef __attribute__((ext_vector_type(8))) float v8f;

#define N_NODES   50000
#define N_EDGES   1600000
#define E_TOT     (N_EDGES + N_NODES)   // edges + self loops
#define N_GRAPHS  512
#define HEADS     4
#define HEAD_DIM  32
#define F         128
#define NEG_SLOPE 0.2f

// ---------- ordered-uint encoding of float for atomicMax over signed floats ----------
__device__ __forceinline__ unsigned ford_enc(float f) {
  unsigned u = __float_as_uint(f);
  return (u & 0x80000000u) ? ~u : (u | 0x80000000u);
}
__device__ __forceinline__ float ford_dec(unsigned u) {
  return (u & 0x80000000u) ? __uint_as_float(u & 0x7FFFFFFFu) : __uint_as_float(~u);
}

__device__ __forceinline__ float leaky(float v) { return v > 0.0f ? v : NEG_SLOPE * v; }

// ---------------------------------------------------------------------------
// GEMM: H[N_NODES,128] = X[N_NODES,128] @ W[128,128] via V_WMMA_F32_16X16X4_F32.
// One wave per 16x16 output tile; block = 8 waves covering one 16x128 row band.
// A frag (16x4 f32): lanes 0-15 -> M=lane, K={k0,k0+1}; lanes 16-31 -> K={k0+2,k0+3}
// B frag (4x16 f32): lanes 0-15 -> N=lane, K={k0,k0+1}; lanes 16-31 -> K={k0+2,k0+3}
// D (16x16 f32):     VGPR r -> M = r + 8*(lane>=16), N = lane&15
// ---------------------------------------------------------------------------
__global__ __launch_bounds__(256) void gat_gemm_wmma(
    const float* __restrict__ X, const float* __restrict__ W, float* __restrict__ H)
{
  const int lane = threadIdx.x & 31;
  const int wave = threadIdx.x >> 5;      // 8 waves -> 8 N-tiles (N = 128)
  const int m0   = blockIdx.x * 16;       // 3125 blocks, M = 50000 exactly
  const int n0   = wave * 16;
  const int half = lane >> 4;
  const int lr   = lane & 15;

  v8f acc = {};
  const float* xrow = X + (size_t)(m0 + lr) * F;
#pragma unroll 8
  for (int k0 = 0; k0 < F; k0 += 4) {
    const int ka = k0 + half * 2;
    v2f a = *(const v2f*)(xrow + ka);           // 8B-aligned (ka even)
    v2f b;
    b.x = W[(size_t)ka * F + n0 + lr];
    b.y = W[(size_t)(ka + 1) * F + n0 + lr];
    acc = __builtin_amdgcn_wmma_f32_16x16x4_f32(
        /*neg_a=*/false, a, /*neg_b=*/false, b,
        /*c_mod=*/(short)0, acc, /*reuse_a=*/false, /*reuse_b=*/false);
  }
#pragma unroll
  for (int r = 0; r < 8; ++r)
    H[(size_t)(m0 + half * 8 + r) * F + n0 + lr] = acc[r];
}

// ---------------------------------------------------------------------------
// Per-(node,head) attention halves: alpha_src/dst = <h[n,h,:], att[h,:]>
// ---------------------------------------------------------------------------
__global__ void gat_alpha(const float* __restrict__ H,
                          const float* __restrict__ a_s, const float* __restrict__ a_d,
                          float* __restrict__ asrc, float* __restrict__ adst)
{
  int t = blockIdx.x * blockDim.x + threadIdx.x;
  if (t >= N_NODES * HEADS) return;
  const int head = t & 3;
  const v4f* hp = (const v4f*)(H + (size_t)(t >> 2) * F + head * HEAD_DIM);
  const v4f* as = (const v4f*)(a_s + head * HEAD_DIM);
  const v4f* ad = (const v4f*)(a_d + head * HEAD_DIM);
  float ss = 0.f, sd = 0.f;
#pragma unroll
  for (int i = 0; i < HEAD_DIM / 4; ++i) {
    v4f v = hp[i], s = as[i], d = ad[i];
    ss += v.x * s.x + v.y * s.y + v.z * s.z + v.w * s.w;
    sd += v.x * d.x + v.y * d.y + v.z * d.z + v.w * d.w;
  }
  asrc[t] = ss;
  adst[t] = sd;
}

// ---------------------------------------------------------------------------
// Edge pass 1: segment max of leaky_relu logits per (dst, head)
// ---------------------------------------------------------------------------
__global__ void gat_edge_max(const int* __restrict__ ei,
                             const float* __restrict__ asrc, const float* __restrict__ adst,
                             unsigned* __restrict__ m_u)
{
  int t = blockIdx.x * blockDim.x + threadIdx.x;
  if (t >= E_TOT * HEADS) return;
  const int e = t >> 2, h = t & 3;
  int s, d;
  if (e < N_EDGES) { s = ei[e]; d = ei[N_EDGES + e]; } else { s = d = e - N_EDGES; }
  const float v = leaky(asrc[s * HEADS + h] + adst[d * HEADS + h]);
  atomicMax(&m_u[d * HEADS + h], ford_enc(v));
}

// ---------------------------------------------------------------------------
// Edge pass 2: denom = segment sum of exp(logit - max) per (dst, head)
// ---------------------------------------------------------------------------
__global__ void gat_edge_sum(const int* __restrict__ ei,
                             const float* __restrict__ asrc, const float* __restrict__ adst,
                             const unsigned* __restrict__ m_u, float* __restrict__ denom)
{
  int t = blockIdx.x * blockDim.x + threadIdx.x;
  if (t >= E_TOT * HEADS) return;
  const int e = t >> 2, h = t & 3;
  int s, d;
  if (e < N_EDGES) { s = ei[e]; d = ei[N_EDGES + e]; } else { s = d = e - N_EDGES; }
  const float v = leaky(asrc[s * HEADS + h] + adst[d * HEADS + h]);
  atomicAdd(&denom[d * HEADS + h], __expf(v - ford_dec(m_u[d * HEADS + h])));
}

// ---------------------------------------------------------------------------
// Edge pass 3: out[dst] += alpha * h[src]. One wave32 per edge; each lane owns
// channel `lane` of each head -> coalesced 128B atomic burst per head.
// ---------------------------------------------------------------------------
__global__ void gat_edge_scatter(const int* __restrict__ ei, const float* __restrict__ H,
                                 const float* __restrict__ asrc, const float* __restrict__ adst,
                                 const unsigned* __restrict__ m_u, const float* __restrict__ denom,
                                 float* __restrict__ out)
{
  const int wid  = (blockIdx.x * blockDim.x + threadIdx.x) >> 5;
  const int lane = threadIdx.x & 31;
  if (wid >= E_TOT) return;
  int s, d;
  if (wid < N_EDGES) { s = ei[wid]; d = ei[N_EDGES + wid]; } else { s = d = wid - N_EDGES; }
  const float* hs = H + (size_t)s * F;
  float* od = out + (size_t)d * F;
#pragma unroll
  for (int h = 0; h < HEADS; ++h) {
    const float v = leaky(asrc[s * HEADS + h] + adst[d * HEADS + h]);
    const float a = __expf(v - ford_dec(m_u[d * HEADS + h])) /
                    (denom[d * HEADS + h] + 1e-16f);
    atomicAdd(&od[h * HEAD_DIM + lane], hs[h * HEAD_DIM + lane] * a);
  }
}

// ---------------------------------------------------------------------------
// bias + ELU (in place); result becomes next layer's input
// ---------------------------------------------------------------------------
__global__ void gat_bias_elu(float* __restrict__ y, const float* __restrict__ b)
{
  int t = blockIdx.x * blockDim.x + threadIdx.x;
  if (t >= N_NODES * F) return;
  const float v = y[t] + b[t & (F - 1)];
  y[t] = v > 0.0f ? v : (__expf(v) - 1.0f);
}

// ---------------------------------------------------------------------------
// Global mean pool over graphs
// ---------------------------------------------------------------------------
__global__ void gat_pool_sum(const float* __restrict__ x, const int* __restrict__ batch,
                             float* __restrict__ out, float* __restrict__ cnt)
{
  int t = blockIdx.x * blockDim.x + threadIdx.x;
  if (t >= N_NODES * F) return;
  const int n = t >> 7, c = t & (F - 1);
  const int g = batch[n];
  atomicAdd(&out[(size_t)g * F + c], x[t]);
  if (c == 0) atomicAdd(&cnt[g], 1.0f);
}

__global__ void gat_pool_div(float* __restrict__ out, const float* __restrict__ cnt)
{
  int t = blockIdx.x * blockDim.x + threadIdx.x;
  if (t >= N_GRAPHS * F) return;
  out[t] /= fmaxf(cnt[t >> 7], 1.0f);
}

// ---------------------------------------------------------------------------
extern "C" void kernel_launch(void* const* d_in, const int* in_sizes, int n_in,
                              void* d_out, int out_size, void* d_ws, size_t ws_size,
                              hipStream_t stream)
{
  (void)in_sizes; (void)n_in; (void)out_size; (void)ws_size;
  const float* x    = (const float*)d_in[0];   // [50000,128]
  const float* Ws   = (const float*)d_in[1];   // [3,128,128]
  const float* a_s  = (const float*)d_in[2];   // [3,4,32]
  const float* a_d  = (const float*)d_in[3];   // [3,4,32]
  const float* bias = (const float*)d_in[4];   // [3,128]
  const int*   ei   = (const int*)d_in[5];     // [2,1600000]
  const int*   bat  = (const int*)d_in[6];     // [50000]
  float* out = (float*)d_out;                  // [512,128]

  // workspace carve-up (~80 MB total)
  char* w = (char*)d_ws;
  const size_t NF = (size_t)N_NODES * F * sizeof(float);       // 25.6 MB
  const size_t NH = (size_t)N_NODES * HEADS * sizeof(float);   // 0.8 MB
  float*    buf0  = (float*)(w);
  float*    buf1  = (float*)(w + NF);
  float*    hbuf  = (float*)(w + 2 * NF);
  float*    asrc  = (float*)(w + 3 * NF);
  float*    adst  = (float*)(w + 3 * NF + NH);
  unsigned* m_u   = (unsigned*)(w + 3 * NF + 2 * NH);
  float*    denom = (float*)(w + 3 * NF + 3 * NH);
  float*    cnt   = (float*)(w + 3 * NF + 4 * NH);

  const int eThreads = E_TOT * HEADS;              // 6.6M
  const int sThreads = E_TOT * 32;                 // one wave per edge

  for (int l = 0; l < 3; ++l) {
    const float* xin  = (l == 0) ? x : (l == 1 ? buf0 : buf1);
    float*       xout = (l == 1) ? buf1 : buf0;
    const float* Wl   = Ws  + (size_t)l * F * F;
    const float* asl  = a_s + (size_t)l * HEADS * HEAD_DIM;
    const float* adl  = a_d + (size_t)l * HEADS * HEAD_DIM;
    const float* bl   = bias + (size_t)l * F;

    gat_gemm_wmma<<<N_NODES / 16, 256, 0, stream>>>(xin, Wl, hbuf);
    gat_alpha<<<(N_NODES * HEADS + 255) / 256, 256, 0, stream>>>(hbuf, asl, adl, asrc, adst);

    hipMemsetAsync(m_u,   0, NH, stream);
    hipMemsetAsync(denom, 0, NH, stream);
    hipMemsetAsync(xout,  0, NF, stream);

    gat_edge_max<<<(eThreads + 255) / 256, 256, 0, stream>>>(ei, asrc, adst, m_u);
    gat_edge_sum<<<(eThreads + 255) / 256, 256, 0, stream>>>(ei, asrc, adst, m_u, denom);
    gat_edge_scatter<<<(sThreads + 255) / 256, 256, 0, stream>>>(ei, hbuf, asrc, adst, m_u, denom, xout);
    gat_bias_elu<<<(N_NODES * F + 255) / 256, 256, 0, stream>>>(xout, bl);
  }

  hipMemsetAsync(out, 0, (size_t)N_GRAPHS * F * sizeof(float), stream);
  hipMemsetAsync(cnt, 0, (size_t)N_GRAPHS * sizeof(float), stream);
  gat_pool_sum<<<(N_NODES * F + 255) / 256, 256, 0, stream>>>(buf0, bat, out, cnt);
  gat_pool_div<<<(N_GRAPHS * F + 255) / 256, 256, 0, stream>>>(out, cnt);
}